// ConditionalMatrixMultiplication_14671608283710
// MI455X (gfx1250) — compile-verified
//
#include <hip/hip_runtime.h>

// ConditionalMatrixMultiplication on MI455X (gfx1250, wave32, WMMA).
//
// Algorithm (15x less FLOPs than reference):
//  1) S[n][i] = sum_o W[n,i,o] for internal nodes n=0..6        (wsum_kernel)
//  2) route[b]: 7 dots x.S + 3-step tree walk                    (route_kernel)
//  3) bucketize samples by route (32-aligned buckets)            (offsets/scatter)
//  4) out[b] = x[b] @ W[7+route[b]] as a gathered WMMA GEMM      (tree_gemm_kernel)
//
// GEMM: block tile 32(M) x 128(N), K chunks of 64 staged via LDS.
// B is stored in LDS K-pair-interleaved (B_sh[p][col][2], p=K/2) so each
// WMMA B operand {K,K+1} is one aligned ds_load_b64 (no regrouping movs).
// 8 waves/WG, each wave = 2 independent 16x16 WMMA chains
// (V_WMMA_F32_16X16X4_F32).

#define B_SAMPLES 4096
#define K_DIM 2048
#define N_DIM 2048
#define N_ROUTES 8
#define TILE_M 32
#define TILE_N 128
#define TILE_K 64
#define MAX_TILES 136           // <= 4096/32 + 8 buckets of padding
#define A_STRIDE 68             // floats; 16B-aligned rows
#define BPROW 288               // floats per K-pair row: 128 cols * 2 + pad
                                // 288 % 64 == 32 -> lane groups on disjoint banks

typedef __attribute__((ext_vector_type(2))) float v2f;
typedef __attribute__((ext_vector_type(8))) float v8f;

// ---- 1) S[n][i] = sum_o W[n*K*N + i*N + o], n in 0..6 -----------------------
__global__ void __launch_bounds__(256) wsum_kernel(const float* __restrict__ W,
                                                   float* __restrict__ S) {
  int n = blockIdx.x >> 11;          // / 2048
  int i = blockIdx.x & 2047;
  const float4* row = (const float4*)(W + ((size_t)n * K_DIM + i) * N_DIM);
  float s = 0.f;
  for (int j = threadIdx.x; j < N_DIM / 4; j += 256) {
    float4 v = row[j];
    s += (v.x + v.y) + (v.z + v.w);
  }
  __shared__ float red[256];
  red[threadIdx.x] = s;
  __syncthreads();
  for (int off = 128; off > 0; off >>= 1) {
    if (threadIdx.x < off) red[threadIdx.x] += red[threadIdx.x + off];
    __syncthreads();
  }
  if (threadIdx.x == 0) S[n * K_DIM + i] = red[0];
}

// ---- workspace init (counts/cursors zero, perm = -1) ------------------------
__global__ void __launch_bounds__(256) init_kernel(int* __restrict__ perm,
                                                   int* __restrict__ tile_route,
                                                   int* __restrict__ counts,
                                                   int* __restrict__ cursor) {
  int idx = blockIdx.x * 256 + threadIdx.x;
  if (idx < MAX_TILES * TILE_M) perm[idx] = -1;
  if (idx < MAX_TILES) tile_route[idx] = 0;
  if (idx < N_ROUTES) { counts[idx] = 0; cursor[idx] = 0; }
}

// ---- 2) routing: 7 dots + tree walk ----------------------------------------
__global__ void __launch_bounds__(256) route_kernel(const float* __restrict__ x,
                                                    const float* __restrict__ S,
                                                    int* __restrict__ route,
                                                    int* __restrict__ counts) {
  int b = blockIdx.x;
  const float* xr = x + (size_t)b * K_DIM;
  float p[7];
#pragma unroll
  for (int n = 0; n < 7; n++) p[n] = 0.f;
  for (int i = threadIdx.x; i < K_DIM; i += 256) {
    float xv = xr[i];
#pragma unroll
    for (int n = 0; n < 7; n++) p[n] += xv * S[n * K_DIM + i];
  }
  __shared__ float red[7][256];
#pragma unroll
  for (int n = 0; n < 7; n++) red[n][threadIdx.x] = p[n];
  __syncthreads();
  for (int off = 128; off > 0; off >>= 1) {
    if (threadIdx.x < off)
#pragma unroll
      for (int n = 0; n < 7; n++) red[n][threadIdx.x] += red[n][threadIdx.x + off];
    __syncthreads();
  }
  if (threadIdx.x == 0) {
    int node = 0;
#pragma unroll
    for (int s = 0; s < 3; s++) node = 2 * node + 1 + (red[node][0] > 0.f ? 1 : 0);
    int r = node - 7;                 // 0..7, selects W[7+r]
    route[b] = r;
    atomicAdd(&counts[r], 1);
  }
}

// ---- 3) bucket offsets (32-aligned) + tile -> route map ---------------------
__global__ void offsets_kernel(const int* __restrict__ counts,
                               int* __restrict__ base,
                               int* __restrict__ tile_route) {
  if (threadIdx.x == 0 && blockIdx.x == 0) {
    int off = 0;
    for (int r = 0; r < N_ROUTES; r++) {
      base[r] = off;
      int tiles = (counts[r] + TILE_M - 1) / TILE_M;
      for (int t = 0; t < tiles; t++) tile_route[off / TILE_M + t] = r;
      off += tiles * TILE_M;
    }
  }
}

__global__ void __launch_bounds__(256) scatter_kernel(const int* __restrict__ route,
                                                      const int* __restrict__ base,
                                                      int* __restrict__ cursor,
                                                      int* __restrict__ perm) {
  int b = blockIdx.x * 256 + threadIdx.x;
  int r = route[b];
  int pos = atomicAdd(&cursor[r], 1);
  perm[base[r] + pos] = b;
}

// ---- 4) gathered/routed GEMM with V_WMMA_F32_16X16X4_F32 --------------------
__global__ void __launch_bounds__(256)
tree_gemm_kernel(const float* __restrict__ x, const float* __restrict__ W,
                 const int* __restrict__ perm, const int* __restrict__ tile_route,
                 float* __restrict__ out) {
  __shared__ float A_sh[TILE_M * A_STRIDE];            // 32 x 64 (+pad)
  __shared__ float B_sh[(TILE_K / 2) * BPROW];         // 32 K-pairs x 128 cols x 2
  __shared__ int rows[TILE_M];

  const int tid = threadIdx.x;
  const int mt = blockIdx.y;
  const int nbase = blockIdx.x * TILE_N;

  if (tid < TILE_M) rows[tid] = perm[mt * TILE_M + tid];
  __syncthreads();
  if (rows[0] < 0) return;            // bucket pad tile: uniformly empty

  const int r = tile_route[mt];
  const float* Wr = W + (size_t)(7 + r) * K_DIM * N_DIM;

  const int wid = tid >> 5;           // wave 0..7
  const int mg = wid >> 2;            // M group 0/1 (rows mg*16..mg*16+15)
  const int ng = wid & 3;             // N group -> cols ng*32 .. ng*32+31
  const int lane = tid & 31;
  const int g = lane >> 4;            // lane group (K halves / M halves)
  const int ln = lane & 15;

  // A staging: thread -> (row 0..31, 8 floats). Pad rows gather row 0:
  // their outputs are never stored, so the values don't matter.
  const int arow = tid >> 3;
  const int acol = (tid & 7) * 8;
  const long agrow = rows[arow] >= 0 ? rows[arow] : 0;
  const float* aptr = x + agrow * (long)K_DIM + acol;
  // B staging: 4 (K-pair, col-quad) units per thread
  const float* bptr = Wr + nbase;

  v8f acc0 = {}, acc1 = {};
  for (int k0 = 0; k0 < K_DIM; k0 += TILE_K) {
    __syncthreads();                  // prior chunk fully consumed
    {
      // A: 32x64, row-major
      float4 a0 = *(const float4*)(aptr + k0);
      float4 a1 = *(const float4*)(aptr + k0 + 4);
      *(float4*)&A_sh[arow * A_STRIDE + acol] = a0;
      *(float4*)&A_sh[arow * A_STRIDE + acol + 4] = a1;
      // B: 64x128 -> pair-interleaved [p][col][2]
#pragma unroll
      for (int it = 0; it < 4; it++) {
        int lin = tid + it * 256;     // 0..1023
        int p = lin >> 5;             // K-pair 0..31
        int c = (lin & 31) * 4;       // col quad
        float4 g0 = *(const float4*)(bptr + (size_t)(k0 + 2 * p) * N_DIM + c);
        float4 g1 = *(const float4*)(bptr + (size_t)(k0 + 2 * p + 1) * N_DIM + c);
        *(float4*)&B_sh[p * BPROW + c * 2] = make_float4(g0.x, g1.x, g0.y, g1.y);
        *(float4*)&B_sh[p * BPROW + c * 2 + 4] = make_float4(g0.z, g1.z, g0.w, g1.w);
      }
    }
    __syncthreads();                  // LDS chunk visible

    const float* Arow = &A_sh[(mg * 16 + ln) * A_STRIDE];
    const float* Bcol = &B_sh[(ng * 32 + ln) * 2];
#pragma unroll
    for (int kk = 0; kk < TILE_K; kk += 4) {
      // A 16x4 f32: lanes 0-15 hold K={kk,kk+1}; lanes 16-31 K={kk+2,kk+3}
      const int p = (kk >> 1) + g;    // K-pair index for this lane group
      v2f a = *(const v2f*)&Arow[kk + 2 * g];
      v2f b0 = *(const v2f*)&Bcol[p * BPROW];        // cols ng*32+ln
      v2f b1 = *(const v2f*)&Bcol[p * BPROW + 32];   // cols ng*32+16+ln
      acc0 = __builtin_amdgcn_wmma_f32_16x16x4_f32(false, a, false, b0,
                                                   (short)0, acc0, false, false);
      acc1 = __builtin_amdgcn_wmma_f32_16x16x4_f32(false, a, false, b1,
                                                   (short)0, acc1, false, false);
    }
  }

  // C/D 16x16 f32: lane -> N = ln, VGPR v -> M = v + 8*g; scatter via perm
  const int col0 = nbase + ng * 32 + ln;
#pragma unroll
  for (int v = 0; v < 8; v++) {
    int grow = rows[mg * 16 + v + 8 * g];
    if (grow >= 0) {
      out[(size_t)grow * N_DIM + col0] = acc0[v];
      out[(size_t)grow * N_DIM + col0 + 16] = acc1[v];
    }
  }
}

extern "C" void kernel_launch(void* const* d_in, const int* in_sizes, int n_in,
                              void* d_out, int out_size, void* d_ws, size_t ws_size,
                              hipStream_t stream) {
  const float* x = (const float*)d_in[0];   // [4096, 2048] f32
  const float* W = (const float*)d_in[1];   // [15, 2048, 2048] f32
  float* out = (float*)d_out;               // [4096, 2048] f32

  // workspace layout
  float* S = (float*)d_ws;                        // 7*2048 f32
  int* route = (int*)(S + 7 * K_DIM);             // 4096
  int* counts = route + B_SAMPLES;                // 8
  int* base = counts + N_ROUTES;                  // 8
  int* cursor = base + N_ROUTES;                  // 8
  int* tile_route = cursor + N_ROUTES;            // 136
  int* perm = tile_route + MAX_TILES;             // 136*32

  (void)in_sizes; (void)n_in; (void)out_size; (void)ws_size;

  init_kernel<<<(MAX_TILES * TILE_M + 255) / 256, 256, 0, stream>>>(
      perm, tile_route, counts, cursor);
  wsum_kernel<<<7 * K_DIM, 256, 0, stream>>>(W, S);
  route_kernel<<<B_SAMPLES, 256, 0, stream>>>(x, S, route, counts);
  offsets_kernel<<<1, 32, 0, stream>>>(counts, base, tile_route);
  scatter_kernel<<<B_SAMPLES / 256, 256, 0, stream>>>(route, base, cursor, perm);
  tree_gemm_kernel<<<dim3(N_DIM / TILE_N, MAX_TILES), 256, 0, stream>>>(
      x, W, perm, tile_route, out);
}